// InvDense_LU_62543313764384
// MI455X (gfx1250) — compile-verified
//
#include <hip/hip_runtime.h>
#include <hip/hip_bf16.h>

typedef __attribute__((ext_vector_type(2))) float v2f;
typedef __attribute__((ext_vector_type(8))) float v8f;

// ---------------------------------------------------------------------------
// Kernel 1: build kmat = P @ (L @ U) into d_ws, and the scalar -log|det|.
// D=64, trivial cost (one workgroup). L = strict-lower(kernel_l) + I,
// U = strict-upper(kernel_u) + diag(sign*exp(abs_log)).
// -log|det(P L U)| = -sum(abs_log) exactly.
// ---------------------------------------------------------------------------
__global__ void __launch_bounds__(256)
build_kmat(const float* __restrict__ P, const float* __restrict__ Lk,
           const float* __restrict__ Uk, const float* __restrict__ usign,
           const float* __restrict__ ulog, float* __restrict__ kmat,
           float* __restrict__ neg_logdet) {
  __shared__ float Ls[64 * 64];
  __shared__ float Us[64 * 64];
  __shared__ float Ms[64 * 64];
  const int t = threadIdx.x;

  for (int i = t; i < 4096; i += 256) {
    const int r = i >> 6, c = i & 63;
    Ls[i] = (c < r) ? Lk[i] : ((c == r) ? 1.0f : 0.0f);
    Us[i] = (c > r) ? Uk[i] : ((c == r) ? usign[r] * expf(ulog[r]) : 0.0f);
  }
  __syncthreads();

  for (int i = t; i < 4096; i += 256) {     // M = L @ U
    const int r = i >> 6, c = i & 63;
    float s = 0.0f;
    for (int k = 0; k < 64; ++k) s = fmaf(Ls[r * 64 + k], Us[k * 64 + c], s);
    Ms[i] = s;
  }
  __syncthreads();

  for (int i = t; i < 4096; i += 256) {     // kmat = P @ M
    const int r = i >> 6, c = i & 63;
    float s = 0.0f;
    for (int k = 0; k < 64; ++k) s = fmaf(P[r * 64 + k], Ms[k * 64 + c], s);
    kmat[i] = s;
  }

  if (t == 0) {
    float s = 0.0f;
    for (int i = 0; i < 64; ++i) s += ulog[i];
    *neg_logdet = -s;
  }
}

// ---------------------------------------------------------------------------
// Kernel 2: out[N,64] = x[N,64] @ kmat[64,64], fp32 WMMA 16x16x4.
// One wave per 16-row strip; grid-stride over strips. The full 64x64 kernel
// matrix is held in VGPRs as B-fragments (16 K-steps x 4 N-tiles x v2f).
//
// Fragment layouts (ISA 7.12.2, wave32):
//   A 16x4 f32 : VGPR0 = {lanes0-15: K=0 | lanes16-31: K=2}, VGPR1 = K=1|K=3,
//                M = lane%16.
//   B 4x16 f32 : VGPR0 = {K=0 | K=2}, VGPR1 = {K=1 | K=3}, N = lane%16.
//   C/D 16x16  : VGPR v = row v (lanes 0-15) / row v+8 (lanes 16-31), N=lane%16.
// ---------------------------------------------------------------------------
__global__ void __launch_bounds__(256)
invlu_gemm(const float* __restrict__ x, const float* __restrict__ kmat,
           float* __restrict__ out, int nstrips) {
  const int lane  = threadIdx.x & 31;
  const int lhalf = lane >> 4;   // 0: K/row-half 0, 1: K/row-half 1
  const int lcol  = lane & 15;
  const int wave  = (blockIdx.x * (blockDim.x >> 5)) + (threadIdx.x >> 5);
  const int nwave = gridDim.x * (blockDim.x >> 5);

  // Load all B fragments of the 64x64 kernel matrix once (stays in VGPRs).
  v2f b[16][4];
#pragma unroll
  for (int ks = 0; ks < 16; ++ks) {
    const int k0 = ks * 4 + lhalf * 2;        // this lane-half holds rows k0,k0+1
#pragma unroll
    for (int j = 0; j < 4; ++j) {
      v2f t;
      t.x = kmat[(k0 + 0) * 64 + j * 16 + lcol];
      t.y = kmat[(k0 + 1) * 64 + j * 16 + lcol];
      b[ks][j] = t;
    }
  }

  for (int s = wave; s < nstrips; s += nwave) {           // uniform per wave
    const float* xr = x + ((size_t)s * 16 + lcol) * 64 + lhalf * 2;

    v8f acc[4] = {};                                      // zero C
#pragma unroll
    for (int ks = 0; ks < 16; ++ks) {
      const v2f a = *(const v2f*)(xr + ks * 4);           // global_load_b64
#pragma unroll
      for (int j = 0; j < 4; ++j) {
        acc[j] = __builtin_amdgcn_wmma_f32_16x16x4_f32(
            /*neg_a=*/false, a, /*neg_b=*/false, b[ks][j],
            /*c_mod=*/(short)0, acc[j], /*reuse_a=*/false, /*reuse_b=*/false);
      }
    }

    float* orow = out + (size_t)s * 16 * 64;
#pragma unroll
    for (int j = 0; j < 4; ++j) {
#pragma unroll
      for (int v = 0; v < 8; ++v) {
        orow[(size_t)(v + lhalf * 8) * 64 + j * 16 + lcol] = acc[j][v];
      }
    }
  }
}

// Scalar tail for rows not covered by 16-row strips (none for N = 2^21,
// but kept for robustness; launched only when needed).
__global__ void invlu_tail(const float* __restrict__ x,
                           const float* __restrict__ kmat,
                           float* __restrict__ out, int row_start, int n) {
  const int total = (n - row_start) * 64;
  int idx = blockIdx.x * blockDim.x + threadIdx.x;
  if (idx >= total) return;
  const int r = row_start + (idx >> 6);
  const int c = idx & 63;
  float s = 0.0f;
  for (int k = 0; k < 64; ++k) s = fmaf(x[(size_t)r * 64 + k], kmat[k * 64 + c], s);
  out[(size_t)r * 64 + c] = s;
}

extern "C" void kernel_launch(void* const* d_in, const int* in_sizes, int n_in,
                              void* d_out, int out_size, void* d_ws, size_t ws_size,
                              hipStream_t stream) {
  const float* x     = (const float*)d_in[0];
  const float* P     = (const float*)d_in[1];
  const float* Lk    = (const float*)d_in[2];
  const float* Uk    = (const float*)d_in[3];
  const float* usign = (const float*)d_in[4];
  const float* ulog  = (const float*)d_in[5];
  float* out  = (float*)d_out;
  float* kmat = (float*)d_ws;                 // 4096 floats = 16 KB scratch

  const int N = in_sizes[0] / 64;             // rows of x
  build_kmat<<<1, 256, 0, stream>>>(P, Lk, Uk, usign, ulog, kmat,
                                    out + (size_t)N * 64);

  const int nstrips = N / 16;
  if (nstrips > 0) {
    int blocks = (nstrips + 7) / 8;           // 8 waves per 256-thread block
    if (blocks > 2048) blocks = 2048;         // grid-stride covers the rest
    invlu_gemm<<<blocks, 256, 0, stream>>>(x, kmat, out, nstrips);
  }
  const int tail_rows = N - nstrips * 16;
  if (tail_rows > 0) {
    const int total = tail_rows * 64;
    invlu_tail<<<(total + 255) / 256, 256, 0, stream>>>(x, kmat, out,
                                                        nstrips * 16, N);
  }
}